// AMPGCN_25933012533532
// MI455X (gfx1250) — compile-verified
//
#include <hip/hip_runtime.h>
#include <math.h>

typedef __attribute__((ext_vector_type(2))) float v2f;
typedef __attribute__((ext_vector_type(8))) float v8f;

#define HD 48          // 3*F hidden width
#define FDIM 16        // input feature dim
#define NCLS 7

// Full 32-lane xor-16 exchange as one ds_swizzle_b32 (group-of-32 mode,
// xor_mask=0x10, or_mask=0, and_mask=0x1f -> imm 0x401f, classic SWAPX16).
__device__ __forceinline__ float swz16(float x) {
    return __int_as_float(__builtin_amdgcn_ds_swizzle(__float_as_int(x), 0x401f));
}

// ---------------------------------------------------------------------------
// h[n,f] = sum_j x[n,j] * W[f,j] + b[f]      (W: [48,16] row major)
// ---------------------------------------------------------------------------
__global__ __launch_bounds__(256) void embed_kernel(
    const float* __restrict__ x, const float* __restrict__ W,
    const float* __restrict__ b, float* __restrict__ h, int total)
{
    int i = blockIdx.x * blockDim.x + threadIdx.x;
    if (i >= total) return;
    int n = i / HD, f = i % HD;
    const float* xr = x + n * FDIM;
    const float* wr = W + f * FDIM;
    float s = b[f];
#pragma unroll
    for (int j = 0; j < FDIM; ++j) s = fmaf(xr[j], wr[j], s);
    h[i] = s;
}

__global__ __launch_bounds__(256) void zero_kernel(float* __restrict__ p, int n)
{
    int i = blockIdx.x * blockDim.x + threadIdx.x;
    if (i < n) p[i] = 0.0f;
}

// ---------------------------------------------------------------------------
// One wave32 per edge.  Lanes 0-15 hold the 16 dst tokens (queries),
// lanes 16-31 hold the 16 src tokens (keys/values).  ED = 3.
//
// WMMA #1 (V_WMMA_F32_16X16X4_F32):  D1[M=s][N=t] = sum_d k[s][d]*q[t][d]
//   A 16x4 layout: lane(M), VGPR+halfwave(K):  lo half K=0,1 ; hi half K=2,3
//   B 4x16 layout: lane(N), VGPR+halfwave(K):  lo half K=0,1 ; hi half K=2,3
//   D 16x16:       VGPR r, lane L -> M = r + 8*(L/16), N = L%16
// Softmax over s = 8 registers + one swz16 pairing.
// WMMA #2..5: out^T[d][t] = sum_s v[s][d]*attn[t][s], K=s chunked 4x.
// ---------------------------------------------------------------------------
__global__ __launch_bounds__(256) void amp_conv_kernel(
    const float* __restrict__ h,
    const int*   __restrict__ srcIdx,
    const int*   __restrict__ dstIdx,
    const float* __restrict__ Wqkv,   // [3,3,3]
    const float* __restrict__ bqkv,   // [3,3]
    const float* __restrict__ Wo,     // [3,3]
    const float* __restrict__ bo,     // [3]
    float* __restrict__ acc,          // [N,48] pre-zeroed
    int E, int applyRelu)
{
    const int lid  = threadIdx.x & 31;
    const int edge = blockIdx.x * (blockDim.x >> 5) + (threadIdx.x >> 5);
    if (edge >= E) return;                       // wave-uniform exit

    const int t    = lid & 15;                   // token index
    const int half = lid >> 4;                   // 0 = dst/query, 1 = src/kv

    const int sN = srcIdx[edge];
    const int dN = dstIdx[edge];
    const int node = half ? sN : dN;

    const float* xp = h + node * HD + t * 3;
    float x0 = xp[0], x1 = xp[1], x2 = xp[2];
    if (applyRelu) { x0 = fmaxf(x0, 0.f); x1 = fmaxf(x1, 0.f); x2 = fmaxf(x2, 0.f); }

    // Merged projection: lower half computes q (p=0), upper half k (p=1);
    // v (p=2) computed by all lanes (consumed from the upper half).
    // W[p][f][d] at p*9 + f*3 + d.
    float r0, r1, r2, v0, v1, v2;
    {
        float w0 = half ? Wqkv[ 9] : Wqkv[0];
        float w1 = half ? Wqkv[10] : Wqkv[1];
        float w2 = half ? Wqkv[11] : Wqkv[2];
        float w3 = half ? Wqkv[12] : Wqkv[3];
        float w4 = half ? Wqkv[13] : Wqkv[4];
        float w5 = half ? Wqkv[14] : Wqkv[5];
        float w6 = half ? Wqkv[15] : Wqkv[6];
        float w7 = half ? Wqkv[16] : Wqkv[7];
        float w8 = half ? Wqkv[17] : Wqkv[8];
        float b0 = half ? bqkv[3]  : bqkv[0];
        float b1 = half ? bqkv[4]  : bqkv[1];
        float b2 = half ? bqkv[5]  : bqkv[2];
        r0 = b0 + w0*x0 + w1*x1 + w2*x2;
        r1 = b1 + w3*x0 + w4*x1 + w5*x2;
        r2 = b2 + w6*x0 + w7*x1 + w8*x2;
        const float* Wv = Wqkv + 18;
        v0 = bqkv[6] + Wv[0]*x0 + Wv[1]*x1 + Wv[2]*x2;
        v1 = bqkv[7] + Wv[3]*x0 + Wv[4]*x1 + Wv[5]*x2;
        v2 = bqkv[8] + Wv[6]*x0 + Wv[7]*x1 + Wv[8]*x2;
    }

    // ---- WMMA 1: transposed scores  D1[s][t] = k[s] . (SC * q[t]) --------
    // r* = q on lower half, k on upper half; xor-16 swaps the halves.
    const float SC = 0.57735026918962576451f;    // 1/sqrt(3)
    float rx0 = swz16(r0);                       // lower: k[t][0]; upper: q[t][0]
    float rx1 = swz16(r1);
    float rx2 = swz16(r2);                       // upper: q[t][2]
    v2f A, B;
    A.x = half ? r2   : rx0;                     // k rows, K=0,1 | K=2,3(pad)
    A.y = half ? 0.0f : rx1;
    B.x = (half ? rx2 : r0) * SC;                // q cols (scale folded here)
    B.y = half ? 0.0f : (r1 * SC);
    v8f Sv = {};
    Sv = __builtin_amdgcn_wmma_f32_16x16x4_f32(false, A, false, B,
                                               (short)0, Sv, false, false);
    float S[8];
#pragma unroll
    for (int r = 0; r < 8; ++r) S[r] = Sv[r];

    // ---- softmax over s (8 regs + cross-half pairing) --------------------
    float m = fmaxf(S[0], S[1]);
#pragma unroll
    for (int r = 2; r < 8; ++r) m = fmaxf(m, S[r]);
    m = fmaxf(m, swz16(m));
    float ssum = 0.f;
#pragma unroll
    for (int r = 0; r < 8; ++r) { S[r] = __expf(S[r] - m); ssum += S[r]; }
    ssum += swz16(ssum);
    float inv = 1.0f / ssum;
#pragma unroll
    for (int r = 0; r < 8; ++r) S[r] *= inv;
    // S[r] @ lane L = attn[t = L%16][s = r + 8*(L/16)]

    // ---- WMMA 2..5: O[d][t] = sum_s v[s][d] * attn[t][s] -----------------
    v8f O = {};
#pragma unroll
    for (int c = 0; c < 4; ++c) {
        // A chunk: lane M=d, K pair: lo half s=4c,4c+1 ; hi half s=4c+2,4c+3
        const int srcA = 16 + 4 * c + (half << 1);
        float w00 = __shfl(v0, srcA,     32);
        float w01 = __shfl(v1, srcA,     32);
        float w02 = __shfl(v2, srcA,     32);
        float w10 = __shfl(v0, srcA + 1, 32);
        float w11 = __shfl(v1, srcA + 1, 32);
        float w12 = __shfl(v2, srcA + 1, 32);
        v2f Av, Bv;
        Av.x = (t == 0) ? w00 : (t == 1) ? w01 : (t == 2) ? w02 : 0.0f;
        Av.y = (t == 0) ? w10 : (t == 1) ? w11 : (t == 2) ? w12 : 0.0f;
        // B chunk: attn[t][s]; needed value is either the lane's own S reg
        // (when the lane's half matches s/8) or its xor-16 partner's.
        if (c < 2) {            // s_lo = 4c+j < 8 ; s_hi = 4c+j+2 < 8
            Bv.x = half ? swz16(S[4*c + 2]) : S[4*c + 0];
            Bv.y = half ? swz16(S[4*c + 3]) : S[4*c + 1];
        } else {                // s_lo = 4c+j >= 8 ; s_hi = 4c+j+2 >= 8
            Bv.x = half ? S[4*c - 6] : swz16(S[4*c - 8]);
            Bv.y = half ? S[4*c - 5] : swz16(S[4*c - 7]);
        }
        O = __builtin_amdgcn_wmma_f32_16x16x4_f32(false, Av, false, Bv,
                                                  (short)0, O, false, false);
    }

    // lanes 0-15: O[0..2] = out[t][0..2]; apply Wo + bo, scatter-add to dst
    if (half == 0) {
        float o0 = O[0], o1 = O[1], o2 = O[2];
#pragma unroll
        for (int f = 0; f < 3; ++f) {
            float msg = bo[f] + Wo[f*3+0]*o0 + Wo[f*3+1]*o1 + Wo[f*3+2]*o2;
            atomicAdd(&acc[dN * HD + t * 3 + f], msg);
        }
    }
}

// ---------------------------------------------------------------------------
// logits = h @ W_lin^T + b ; out = log_softmax(logits)
// ---------------------------------------------------------------------------
__global__ __launch_bounds__(256) void head_kernel(
    const float* __restrict__ h, const float* __restrict__ W,
    const float* __restrict__ b, float* __restrict__ out, int N)
{
    int n = blockIdx.x * blockDim.x + threadIdx.x;
    if (n >= N) return;
    const float* hr = h + n * HD;
    float logit[NCLS];
    float m = -INFINITY;
#pragma unroll
    for (int c = 0; c < NCLS; ++c) {
        float s = b[c];
        const float* wr = W + c * HD;
#pragma unroll
        for (int j = 0; j < HD; ++j) s = fmaf(hr[j], wr[j], s);
        logit[c] = s;
        m = fmaxf(m, s);
    }
    float se = 0.f;
#pragma unroll
    for (int c = 0; c < NCLS; ++c) se += __expf(logit[c] - m);
    float lse = __logf(se) + m;
#pragma unroll
    for (int c = 0; c < NCLS; ++c) out[n * NCLS + c] = logit[c] - lse;
}

// ---------------------------------------------------------------------------
extern "C" void kernel_launch(void* const* d_in, const int* in_sizes, int n_in,
                              void* d_out, int out_size, void* d_ws, size_t ws_size,
                              hipStream_t stream)
{
    const float* x       = (const float*)d_in[0];
    const int*   ei      = (const int*)  d_in[1];
    const float* W_embed = (const float*)d_in[2];
    const float* b_embed = (const float*)d_in[3];
    const float* Wqkv1   = (const float*)d_in[4];
    const float* bqkv1   = (const float*)d_in[5];
    const float* Wo1     = (const float*)d_in[6];
    const float* bo1     = (const float*)d_in[7];
    const float* Wqkv2   = (const float*)d_in[8];
    const float* bqkv2   = (const float*)d_in[9];
    const float* Wo2     = (const float*)d_in[10];
    const float* bo2     = (const float*)d_in[11];
    const float* W_lin   = (const float*)d_in[12];
    const float* b_lin   = (const float*)d_in[13];

    const int N = in_sizes[0] / FDIM;
    const int E = in_sizes[1] / 2;
    const int* src = ei;
    const int* dst = ei + E;

    float* h1 = (float*)d_ws;          // [N,48]
    float* h2 = h1 + (size_t)N * HD;   // [N,48] conv1 accumulator
    float* h3 = h2 + (size_t)N * HD;   // [N,48] conv2 accumulator

    // zero both accumulators every call (graph-replay safe)
    int nz = 2 * N * HD;
    zero_kernel<<<(nz + 255) / 256, 256, 0, stream>>>(h2, nz);

    int ne = N * HD;
    embed_kernel<<<(ne + 255) / 256, 256, 0, stream>>>(x, W_embed, b_embed, h1, ne);

    const int wavesPerBlock = 8;       // 256 threads = 8 wave32
    int convBlocks = (E + wavesPerBlock - 1) / wavesPerBlock;
    amp_conv_kernel<<<convBlocks, 256, 0, stream>>>(
        h1, src, dst, Wqkv1, bqkv1, Wo1, bo1, h2, E, /*relu=*/0);
    amp_conv_kernel<<<convBlocks, 256, 0, stream>>>(
        h2, src, dst, Wqkv2, bqkv2, Wo2, bo2, h3, E, /*relu=*/1);

    head_kernel<<<(N + 255) / 256, 256, 0, stream>>>(h3, W_lin, b_lin,
                                                     (float*)d_out, N);
}